// EdgeNetwork_24747601560132
// MI455X (gfx1250) — compile-verified
//
#include <hip/hip_runtime.h>

typedef __attribute__((ext_vector_type(2))) float v2f;
typedef __attribute__((ext_vector_type(8))) float v8f;

#define E_TOTAL 65536
#define SEGS    4480
#define KP      1088      // (16 bond feats + 1 bias feat) * 64
#define NSTEP   272       // KP / 4  (K-steps of the 16x16x4 WMMA)

// ---------------------------------------------------------------------------
// Pack B = [ kernel reshaped ; bias ] into per-lane WMMA fragment order.
//   B[c, i] with c = k*64 + j  (k<16 -> kernel[k, i*64+j], k==16 -> bias[i*64+j])
//   Fragment for (k-step s, N-tile t): Bf[(s*4+t)*64 + lane*2 + v]
//     holds B[4s + v + 2*(lane>=16), 16t + (lane&15)]
//   so each lane loads its 2 B-VGPRs with ONE b64 load.
// ---------------------------------------------------------------------------
__global__ __launch_bounds__(256)
void pack_B(const float* __restrict__ kern, const float* __restrict__ bias,
            float* __restrict__ Bf) {
  int idx = blockIdx.x * 256 + threadIdx.x;      // 272*256 = 69632 elements
  int s = idx >> 8;
  int t = (idx >> 6) & 3;
  int q = idx & 63;
  int l = q >> 1;
  int v = q & 1;
  int c = 4 * s + v + 2 * (l >> 4);
  int i = 16 * t + (l & 15);
  int k = c >> 6;
  int j = c & 63;
  Bf[idx] = (k < 16) ? kern[k * 4096 + i * 64 + j] : bias[i * 64 + j];
}

// ---------------------------------------------------------------------------
// Fused edge-message GEMM + scatter-add.
// Block = 128 threads = 4 waves; each wave owns TWO 16-edge M-tiles (32 edges)
// so every B fragment loaded from L2 feeds 2 WMMAs (register-level M reuse).
// messages[32 x 64] = A[32 x 1088] @ B[1088 x 64] via v_wmma_f32_16x16x4_f32,
// A built on the fly as outer(bond', nbr) from LDS.
// ---------------------------------------------------------------------------
__global__ __launch_bounds__(128)
void edge_mp(const float* __restrict__ atom,      // [4480, 64]
             const float* __restrict__ bond,      // [65536, 16]
             const long long* __restrict__ pair,  // [65536, 2] (dst, src)
             const float* __restrict__ Bf,        // packed B fragments
             float* __restrict__ out) {           // [4480, 64], pre-zeroed
  __shared__ float bondS[4][32][20];   // 16 feats + constant-1, padded
  __shared__ float nS[4][32][68];      // gathered neighbor feats, padded
  __shared__ int   dstS[4][32];

  const int tid = threadIdx.x;
  const int w   = tid >> 5;          // wave in block (0..3)
  const int l   = tid & 31;          // lane
  const int m   = l & 15;            // edge row within tile / N within tile
  const int h   = l >> 4;            // lane half

  const int ebase = blockIdx.x * 128 + w * 32;

  // ---- stage per-wave data (2 tiles = 32 edges) into LDS ----------------
  #pragma unroll
  for (int tt = 0; tt < 2; ++tt) {
    const int row = tt * 16 + m;
    const int e   = ebase + row;
    #pragma unroll
    for (int i = 0; i < 8; ++i)
      bondS[w][row][h * 8 + i] = bond[e * 16 + h * 8 + i];
    if (h == 0) {
      bondS[w][row][16] = 1.0f;                    // bias feature
      dstS[w][row] = (int)pair[2 * e];             // destination atom
    }
    const int src = (int)pair[2 * e + 1];
    #pragma unroll
    for (int i = 0; i < 32; ++i)
      nS[w][row][h * 32 + i] = atom[src * 64 + h * 32 + i];
  }

  __syncthreads();

  // ---- main GEMM loop: 272 K-steps, 2 M-tiles x 4 N-tiles ---------------
  v8f a00 = {}, a01 = {}, a02 = {}, a03 = {};   // tile 0 accumulators
  v8f a10 = {}, a11 = {}, a12 = {}, a13 = {};   // tile 1 accumulators
  const int g2 = h * 2;              // K offset of this lane half (A & B layout)
  const int r0 = m;                  // LDS row for tile 0
  const int r1 = 16 + m;             // LDS row for tile 1

  for (int k = 0; k < 17; ++k) {
    const float bk0 = bondS[w][r0][k];
    const float bk1 = bondS[w][r1][k];
    #pragma unroll 4
    for (int jj = 0; jj < 16; ++jj) {
      const int s  = k * 16 + jj;    // K-step index, c = 4s + (0..3)
      const int j0 = 4 * jj + g2;    // neighbor-feature column for this lane
      v2f aA, aB;
      aA.x = bk0 * nS[w][r0][j0];    // A vgpr0: K = 2h
      aA.y = bk0 * nS[w][r0][j0 + 1];// A vgpr1: K = 2h+1
      aB.x = bk1 * nS[w][r1][j0];
      aB.y = bk1 * nS[w][r1][j0 + 1];

      const v2f* bp = (const v2f*)(Bf + (s << 8)) + l;  // (s*4+t)*64 floats
      v2f b0 = bp[0];
      v2f b1 = bp[32];
      v2f b2 = bp[64];
      v2f b3 = bp[96];

      a00 = __builtin_amdgcn_wmma_f32_16x16x4_f32(false, aA, false, b0, (short)0, a00, false, false);
      a10 = __builtin_amdgcn_wmma_f32_16x16x4_f32(false, aB, false, b0, (short)0, a10, false, false);
      a01 = __builtin_amdgcn_wmma_f32_16x16x4_f32(false, aA, false, b1, (short)0, a01, false, false);
      a11 = __builtin_amdgcn_wmma_f32_16x16x4_f32(false, aB, false, b1, (short)0, a11, false, false);
      a02 = __builtin_amdgcn_wmma_f32_16x16x4_f32(false, aA, false, b2, (short)0, a02, false, false);
      a12 = __builtin_amdgcn_wmma_f32_16x16x4_f32(false, aB, false, b2, (short)0, a12, false, false);
      a03 = __builtin_amdgcn_wmma_f32_16x16x4_f32(false, aA, false, b3, (short)0, a03, false, false);
      a13 = __builtin_amdgcn_wmma_f32_16x16x4_f32(false, aB, false, b3, (short)0, a13, false, false);
    }
  }

  // ---- scatter-add using documented C/D layout --------------------------
  // VGPR r: lanes 0-15 -> M=r, lanes 16-31 -> M=r+8; N = lane & 15.
  #pragma unroll
  for (int r = 0; r < 8; ++r) {
    const int row = r + 8 * h;
    float* op0 = out + dstS[w][row] * 64 + m;
    atomicAdd(op0 +  0, a00[r]);
    atomicAdd(op0 + 16, a01[r]);
    atomicAdd(op0 + 32, a02[r]);
    atomicAdd(op0 + 48, a03[r]);
    float* op1 = out + dstS[w][16 + row] * 64 + m;
    atomicAdd(op1 +  0, a10[r]);
    atomicAdd(op1 + 16, a11[r]);
    atomicAdd(op1 + 32, a12[r]);
    atomicAdd(op1 + 48, a13[r]);
  }
}

// ---------------------------------------------------------------------------
extern "C" void kernel_launch(void* const* d_in, const int* in_sizes, int n_in,
                              void* d_out, int out_size, void* d_ws, size_t ws_size,
                              hipStream_t stream) {
  const float*     atom = (const float*)d_in[0];      // [4480, 64]
  const float*     bond = (const float*)d_in[1];      // [65536, 16]
  const long long* pair = (const long long*)d_in[2];  // [65536, 2]
  const float*     kern = (const float*)d_in[3];      // [16, 4096]
  const float*     bias = (const float*)d_in[4];      // [4096]
  float* Bf  = (float*)d_ws;                          // 272*256 floats = 278528 B
  float* out = (float*)d_out;

  hipMemsetAsync(out, 0, (size_t)out_size * sizeof(float), stream);
  pack_B<<<NSTEP, 256, 0, stream>>>(kern, bias, Bf);
  edge_mp<<<E_TOTAL / 128, 128, 0, stream>>>(atom, bond, pair, Bf, out);
}